// GCNRecsys_14748917694874
// MI455X (gfx1250) — compile-verified
//
#include <hip/hip_runtime.h>

// ---------------- problem constants (from the reference) ----------------
#define N_USERS  100000
#define N_ITEMS  50000
#define N_NODES  (N_USERS + N_ITEMS)   // 150000
#define NNZ_CT   1500000
#define BATCH_CT 1048576
#define DIM      64

typedef __attribute__((ext_vector_type(2))) float v2f;
typedef __attribute__((ext_vector_type(8))) float v8f;

// ---------------- zero the two H buffers in workspace ----------------
__global__ __launch_bounds__(256) void zero_kernel(float4* __restrict__ p, int n4) {
    int i = blockIdx.x * blockDim.x + threadIdx.x;
    if (i < n4) p[i] = make_float4(0.f, 0.f, 0.f, 0.f);
}

// ---------------- COO SpMM: Hout[r,:] += v * Hin[c,:] ----------------
// 16 threads per edge, each thread owns a 4-float quad (float4 gather,
// 4 non-returning f32 global atomics). Source row comes from srcA if
// c < split else srcB (lets layer 1 read Eu/Ei without materializing X0).
__global__ __launch_bounds__(256) void spmm_kernel(
    const int*   __restrict__ rows, const int* __restrict__ cols,
    const float* __restrict__ vals,
    const float* __restrict__ srcA, const float* __restrict__ srcB, int split,
    float* __restrict__ Hout, int nnz) {
    long long t = (long long)blockIdx.x * blockDim.x + threadIdx.x;
    int e = (int)(t >> 4);
    if (e >= nnz) return;
    int dq = ((int)t & 15) << 2;          // dim quad offset: 0,4,...,60
    int r = rows[e];
    int c = cols[e];
    float v = vals[e];
    const float* src = (c < split) ? (srcA + (size_t)c * DIM)
                                   : (srcB + (size_t)(c - split) * DIM);
    float4 h = *(const float4*)(src + dq);
    float* dst = Hout + (size_t)r * DIM + dq;
    unsafeAtomicAdd(dst + 0, v * h.x);    // global_atomic_add_f32, no return
    unsafeAtomicAdd(dst + 1, v * h.y);
    unsafeAtomicAdd(dst + 2, v * h.z);
    unsafeAtomicAdd(dst + 3, v * h.w);
}

// ---------------- batched dot products via f32 WMMA ----------------
// One wave32 per tile of 16 (u,i) pairs. D = U_tile(16x64) x I_tile(64x16)^T
// chained as 16x V_WMMA_F32_16X16X4_F32 (full f32 precision); the diagonal
// of D holds the 16 dot products.
// A layout (16x4 f32): lane&15 = M, VGPR0/1 hold K = 2*(lane>>4) + {0,1}.
// B layout (4x16 f32): symmetric with lane&15 = N.  Both are one b64 load.
__global__ __launch_bounds__(256) void pred_kernel(
    const float* __restrict__ U, const float* __restrict__ I,
    const int*   __restrict__ u_idx, const int* __restrict__ i_idx,
    const float* __restrict__ bu, const float* __restrict__ bi,
    const float* __restrict__ mu, float* __restrict__ out) {
    int wave = blockIdx.x * (blockDim.x >> 5) + (threadIdx.x >> 5);
    int lane = threadIdx.x & 31;
    int base = wave << 4;                 // first pair of this tile
    int m    = lane & 15;                 // row (A) / col (B) index in tile
    int hi   = lane >> 4;                 // K-half selector
    int ur = u_idx[base + m];
    int ir = i_idx[base + m];
    const float* up = U + (size_t)ur * DIM;
    const float* ip = I + (size_t)ir * DIM;

    v8f c = {};
#pragma unroll
    for (int k0 = 0; k0 < DIM; k0 += 4) {
        v2f a = *(const v2f*)(up + k0 + 2 * hi);
        v2f b = *(const v2f*)(ip + k0 + 2 * hi);
        // (neg_a, A, neg_b, B, c_mod, C, reuse_a, reuse_b)
        c = __builtin_amdgcn_wmma_f32_16x16x4_f32(
                false, a, false, b, (short)0, c, false, false);
    }

    // Extract diag(D): element (j,j) lives in C-VGPR (j&7) at lane j (j<8)
    // or lane j+16 (j>=8).  Broadcast with shuffles; lane j keeps diag(j).
    float d = 0.f;
#pragma unroll
    for (int j = 0; j < 8; ++j) {
        float vlo = __shfl(c[j], j, 32);
        float vhi = __shfl(c[j], j + 24, 32);
        if (lane == j)     d = vlo;
        if (lane == j + 8) d = vhi;
    }

    if (lane < 16) {
        out[base + lane] = d + bu[ur] + bi[ir] + mu[0];
    }
}

// ---------------- host-side launch ----------------
extern "C" void kernel_launch(void* const* d_in, const int* in_sizes, int n_in,
                              void* d_out, int out_size, void* d_ws, size_t ws_size,
                              hipStream_t stream) {
    const int*   u_idx  = (const int*)d_in[0];
    const int*   i_idx  = (const int*)d_in[1];
    const int*   A_rows = (const int*)d_in[2];
    const int*   A_cols = (const int*)d_in[3];
    const float* A_vals = (const float*)d_in[4];
    const float* Eu     = (const float*)d_in[5];
    const float* Ei     = (const float*)d_in[6];
    const float* bu     = (const float*)d_in[7];
    const float* bi     = (const float*)d_in[8];
    const float* mu     = (const float*)d_in[9];
    float* out = (float*)d_out;

    float* H1 = (float*)d_ws;                       // 150000*64 f32 = 38.4 MB
    float* H2 = H1 + (size_t)N_NODES * DIM;         // another 38.4 MB

    // zero H1 and H2
    int n4 = 2 * N_NODES * DIM / 4;
    zero_kernel<<<(n4 + 255) / 256, 256, 0, stream>>>((float4*)d_ws, n4);

    // layer 1: H1 = A @ concat(Eu, Ei)
    long long spmm_threads = (long long)NNZ_CT * 16;
    int spmm_blocks = (int)((spmm_threads + 255) / 256);
    spmm_kernel<<<spmm_blocks, 256, 0, stream>>>(
        A_rows, A_cols, A_vals, Eu, Ei, N_USERS, H1, NNZ_CT);

    // layer 2: H2 = A @ H1  (split >= N_NODES so srcA==H1 always taken)
    spmm_kernel<<<spmm_blocks, 256, 0, stream>>>(
        A_rows, A_cols, A_vals, H1, H1, N_NODES + 1, H2, NNZ_CT);

    // predictions: U = H2[:N_USERS], I = H2[N_USERS:]
    // 65536 tiles of 16 pairs, 8 waves per 256-thread block -> 8192 blocks
    int pred_blocks = (BATCH_CT / 16) / 8;
    pred_kernel<<<pred_blocks, 256, 0, stream>>>(
        H2, H2 + (size_t)N_USERS * DIM, u_idx, i_idx, bu, bi, mu, out);
}